// MLCTemporal_75325136437730
// MI455X (gfx1250) — compile-verified
//
#include <hip/hip_runtime.h>
#include <hip/hip_bf16.h>

// ---- problem dims -----------------------------------------------------------
#define B_ROWS   4096
#define D_SAE    16384
#define KDIM     2304      // L * D_IN = 3*768
#define ROW_DEC  4608      // T * L * D_IN = 2*3*768
#define TOPK     64

typedef float v2f __attribute__((ext_vector_type(2)));
typedef float v8f __attribute__((ext_vector_type(8)));
typedef unsigned int u32x4 __attribute__((ext_vector_type(4)));
typedef int i32x8 __attribute__((ext_vector_type(8)));
typedef int i32x4 __attribute__((ext_vector_type(4)));

#if defined(__has_builtin)
#if __has_builtin(__builtin_amdgcn_tensor_load_to_lds) && __has_builtin(__builtin_amdgcn_s_wait_tensorcnt)
#define HAVE_TDM 1
#endif
#endif

__device__ __forceinline__ unsigned mapkey(float f) {
  unsigned u = __float_as_uint(f);
  return (u & 0x80000000u) ? ~u : (u | 0x80000000u);  // monotone order-preserving
}

// ---- K0: xsum[b, l*768+d] = x[b,0,l,d] + x[b,1,l,d] ------------------------
__global__ __launch_bounds__(256) void sae_xsum_kernel(
    const float* __restrict__ x, float* __restrict__ xsum) {
  size_t i = ((size_t)blockIdx.x * 256 + threadIdx.x) * 4;   // float4 index
  size_t b = i / KDIM;
  size_t r = i % KDIM;                                       // KDIM % 4 == 0
  const float4 x0 = *(const float4*)(x + b * ROW_DEC + r);
  const float4 x1 = *(const float4*)(x + b * ROW_DEC + KDIM + r);
  float4 s;
  s.x = x0.x + x1.x; s.y = x0.y + x1.y; s.z = x0.z + x1.z; s.w = x0.w + x1.w;
  *(float4*)(xsum + i) = s;
}

// ---- K1: pre = xsum @ W_enc + b_enc  (fp32 WMMA 16x16x4, TDM-fed) ----------
#define BM 32
#define BN 256
#define BK 32
#define A_STRIDE 40      // As[k][m], padded: 2*40=80 ≡ 16 mod 64 banks
#define B_STRIDE 264     // Bs[k][n], padded: 2*264=528 ≡ 16 mod 64 banks
#define A_BYTES  (BK * A_STRIDE * 4)   //  5120 B
#define B_BYTES  (BK * B_STRIDE * 4)   // 33792 B
#define GEMM_LDS (2 * A_BYTES + 2 * B_BYTES)  // 77824 B, double-buffered

#ifdef HAVE_TDM
// Issue one TDM descriptor: load a 32x256 f32 tile (row stride 16384 elems)
// into LDS at lds_off, inserting 8 DWORDs of padding after every 256 DWORDs
// (one row) -> LDS row stride 264 floats == B_STRIDE.
__device__ __forceinline__ void tdm_load_b_tile(const float* gsrc, unsigned lds_off) {
  const unsigned long long ga = (unsigned long long)(uintptr_t)gsrc;
  u32x4 g0;
  g0.x = 1u;                                                  // count=1, user D#
  g0.y = lds_off;                                             // lds_addr (bytes)
  g0.z = (unsigned)(ga & 0xFFFFFFFFu);                        // global_addr[31:0]
  g0.w = (unsigned)((ga >> 32) & 0x01FFFFFFu) | 0x80000000u;  // addr[56:32] | type=2
  i32x8 g1;
  g1[0] = (2 << 16) | (1 << 20) | (7 << 22) | (7 << 25);      // data_size=4B, pad_en,
                                                              // interval=256dw, amount=8dw
  g1[1] = (int)(16384u << 16);   // tensor_dim0 = 16384 (bits 79:48, low half here)
  g1[2] = (int)(0xFFFFu << 16);  // tensor_dim0 hi = 0; tensor_dim1 = 0xFFFF (ample)
  g1[3] = (int)(256u  << 16);    // tensor_dim1 hi = 0; tile_dim0 = 256
  g1[4] = 32;                    // tile_dim1 = 32; tile_dim2 = 0
  g1[5] = 16384;                 // tensor_dim0_stride (elements)
  g1[6] = 0;                     // stride hi; tensor_dim1_stride lo
  g1[7] = 0;
  const i32x4 gz4 = {0, 0, 0, 0};
  const i32x8 gz8 = {0, 0, 0, 0, 0, 0, 0, 0};
  __builtin_amdgcn_tensor_load_to_lds(g0, g1, gz4, gz4, gz8, 0);
}
#endif

__device__ __forceinline__ void stage_a_tile(float* __restrict__ As,
                                             const float* __restrict__ xsum,
                                             int m0, int kb, int tid) {
  const int r  = tid >> 3;           // m row 0..31
  const int c4 = (tid & 7) << 2;     // k col 0..28
  const float4 av = *(const float4*)(xsum + (size_t)(m0 + r) * KDIM + kb + c4);
  As[(c4 + 0) * A_STRIDE + r] = av.x;
  As[(c4 + 1) * A_STRIDE + r] = av.y;
  As[(c4 + 2) * A_STRIDE + r] = av.z;
  As[(c4 + 3) * A_STRIDE + r] = av.w;
}

#ifndef HAVE_TDM
__device__ __forceinline__ void stage_b_tile(float* __restrict__ Bs,
                                             const float* __restrict__ W_enc,
                                             int n0, int kb, int tid) {
  #pragma unroll
  for (int it = 0; it < 8; it++) {
    const int idx = tid + it * 256;        // 0..2047 float4 slots
    const int r   = idx >> 6;              // k row 0..31
    const int c4  = (idx & 63) << 2;       // n col 0..252
    const float4 bv = *(const float4*)(W_enc + (size_t)(kb + r) * D_SAE + n0 + c4);
    *(float4*)(Bs + r * B_STRIDE + c4) = bv;
  }
}
#endif

__device__ __forceinline__ void mma_tile(const float* __restrict__ As,
                                         const float* __restrict__ Bs,
                                         v8f acc[2][2], int wn_off,
                                         int mloc, int khalf) {
  #pragma unroll
  for (int k4 = 0; k4 < BK; k4 += 4) {
    v2f a[2], b[2];
    #pragma unroll
    for (int m = 0; m < 2; m++) {
      a[m].x = As[(k4 + 2 * khalf    ) * A_STRIDE + m * 16 + mloc];
      a[m].y = As[(k4 + 2 * khalf + 1) * A_STRIDE + m * 16 + mloc];
    }
    #pragma unroll
    for (int n = 0; n < 2; n++) {
      const int nn = wn_off + n * 16 + mloc;
      b[n].x = Bs[(k4 + 2 * khalf    ) * B_STRIDE + nn];
      b[n].y = Bs[(k4 + 2 * khalf + 1) * B_STRIDE + nn];
    }
    #pragma unroll
    for (int m = 0; m < 2; m++)
      #pragma unroll
      for (int n = 0; n < 2; n++)
        acc[m][n] = __builtin_amdgcn_wmma_f32_16x16x4_f32(
            false, a[m], false, b[n], (short)0, acc[m][n], false, false);
  }
}

__global__ __launch_bounds__(256) void sae_enc_gemm_kernel(
    const float* __restrict__ xsum, const float* __restrict__ W_enc,
    const float* __restrict__ b_enc, float* __restrict__ pre) {
  extern __shared__ char gsm[];
  // LDS layout: [ As buf0 | As buf1 | Bs buf0 | Bs buf1 ]
  // (no pointer arrays: addrspacecast in aggregate initializers doesn't link)

  const int tid  = threadIdx.x;
  const int lane = tid & 31;
  const int wid  = tid >> 5;            // 8 waves, each owns 32 N columns
  const int m0   = blockIdx.y * BM;     // 128 M blocks
  const int n0   = blockIdx.x * BN;     // 64  N blocks
  const int mloc  = lane & 15;
  const int khalf = lane >> 4;
  const int wn_off = wid * 32;

  v8f acc[2][2] = {};

  // ---- prologue: stage tile 0 into buffer 0 ----
#ifdef HAVE_TDM
  if (wid == 0) tdm_load_b_tile(W_enc + n0, 2u * A_BYTES);
#else
  stage_b_tile((float*)(gsm + 2 * A_BYTES), W_enc, n0, 0, tid);
#endif
  stage_a_tile((float*)gsm, xsum, m0, 0, tid);
#ifdef HAVE_TDM
  if (wid == 0) __builtin_amdgcn_s_wait_tensorcnt(0);
#endif
  __syncthreads();

  const int nk = KDIM / BK;             // 72
  for (int it = 0; it < nk; ++it) {
    const int cur = it & 1;
    const int nxt = cur ^ 1;
    float* As_cur = (float*)(gsm + cur * A_BYTES);
    float* As_nxt = (float*)(gsm + nxt * A_BYTES);
    float* Bs_cur = (float*)(gsm + 2 * A_BYTES + cur * B_BYTES);
    const unsigned bs_nxt_off = 2u * A_BYTES + (unsigned)nxt * B_BYTES;
    const int kbn = (it + 1) * BK;
    if (kbn < KDIM) {                   // prefetch next tile (overlaps compute)
#ifdef HAVE_TDM
      if (wid == 0) tdm_load_b_tile(W_enc + (size_t)kbn * D_SAE + n0, bs_nxt_off);
#else
      stage_b_tile((float*)(gsm + bs_nxt_off), W_enc, n0, kbn, tid);
#endif
      stage_a_tile(As_nxt, xsum, m0, kbn, tid);
    }
    mma_tile(As_cur, Bs_cur, acc, wn_off, mloc, khalf);
#ifdef HAVE_TDM
    if (wid == 0) __builtin_amdgcn_s_wait_tensorcnt(0);
#endif
    __syncthreads();
  }

  // epilogue: + b_enc, store pre (scratch in z region of d_out)
  #pragma unroll
  for (int n = 0; n < 2; n++) {
    const int col = n0 + wn_off + n * 16 + mloc;
    const float be = b_enc[col];
    #pragma unroll
    for (int m = 0; m < 2; m++) {
      #pragma unroll
      for (int v = 0; v < 8; v++) {
        const int row = m0 + m * 16 + khalf * 8 + v;
        pre[(size_t)row * D_SAE + col] = acc[m][n][v] + be;
      }
    }
  }
}

// ---- K2: per-row exact top-64 (radix select) + relu + compact list ---------
__global__ __launch_bounds__(256) void sae_topk_kernel(
    float* __restrict__ zbuf,            // in: pre, out: z (same buffer)
    int* __restrict__ cnt_out, int* __restrict__ idx_out,
    float* __restrict__ val_out) {
  extern __shared__ char smem[];
  float* sval  = (float*)smem;                         // 16384 floats
  int*   hist  = (int*)(smem + D_SAE * sizeof(float)); // 256 ints
  int*   sscan = hist + 256;                           // 256 ints
  __shared__ unsigned sb_prefix;
  __shared__ int sb_want;

  const int tid = threadIdx.x;
  const int row = blockIdx.x;
  float* rowp = zbuf + (size_t)row * D_SAE;

  for (int i = tid; i < D_SAE; i += 256) sval[i] = rowp[i];
  __syncthreads();

  // 4-pass radix select for the 64th-largest key
  unsigned prefix = 0;
  int want = TOPK;
  for (int pass = 0; pass < 4; pass++) {
    hist[tid] = 0;
    __syncthreads();
    const int shift = 24 - 8 * pass;
    for (int i = tid; i < D_SAE; i += 256) {
      const unsigned key = mapkey(sval[i]);
      const bool ok = (pass == 0) || ((key >> (shift + 8)) == prefix);
      if (ok) atomicAdd(&hist[(key >> shift) & 255], 1);
    }
    __syncthreads();
    if (tid == 0) {
      int c = 0, b = 255;
      for (; b >= 0; b--) { c += hist[b]; if (c >= want) break; }
      sb_prefix = (prefix << 8) | (unsigned)b;
      sb_want = want - (c - hist[b]);
    }
    __syncthreads();
    prefix = sb_prefix;
    want = sb_want;
    __syncthreads();
  }
  const unsigned kth = prefix;   // exact 64th-largest key
  const int r = want;            // # of ties at kth to keep (index order)

  // contiguous chunk per thread: [tid*64, tid*64+64)
  const int base = tid * 64;
  int myEq = 0;
  for (int j = 0; j < 64; j++)
    if (mapkey(sval[base + j]) == kth) myEq++;
  sscan[tid] = myEq;
  __syncthreads();
  if (tid == 0) {
    int s = 0;
    for (int t2 = 0; t2 < 256; t2++) { int c = sscan[t2]; sscan[t2] = s; s += c; }
  }
  __syncthreads();
  const int eqBase = sscan[tid];
  __syncthreads();

  // select, relu, write z into LDS in place; count keepers (val > 0)
  int myKeep = 0, e = 0;
  for (int j = 0; j < 64; j++) {
    const float v = sval[base + j];
    const unsigned key = mapkey(v);
    bool sel;
    if (key > kth)        sel = true;
    else if (key == kth)  { sel = (eqBase + e) < r; e++; }
    else                  sel = false;
    const float zv = (sel && v > 0.0f) ? v : 0.0f;
    sval[base + j] = zv;
    if (zv > 0.0f) myKeep++;
  }
  sscan[tid] = myKeep;
  __syncthreads();
  if (tid == 0) {
    int s = 0;
    for (int t2 = 0; t2 < 256; t2++) { int c = sscan[t2]; sscan[t2] = s; s += c; }
    cnt_out[row] = s;
  }
  __syncthreads();
  int kpos = sscan[tid];
  for (int j = 0; j < 64; j++) {
    const float zv = sval[base + j];
    if (zv > 0.0f) {
      idx_out[(size_t)row * TOPK + kpos] = base + j;
      val_out[(size_t)row * TOPK + kpos] = zv;
      kpos++;
    }
  }
  __syncthreads();
  for (int i = tid; i < D_SAE; i += 256) rowp[i] = sval[i];   // coalesced z write
}

// ---- K3: sparse decode + per-row SSE ---------------------------------------
__global__ __launch_bounds__(256) void sae_decode_kernel(
    const float* __restrict__ x, const float* __restrict__ W_dec,
    const float* __restrict__ b_dec, const int* __restrict__ cnt,
    const int* __restrict__ idxl, const float* __restrict__ vall,
    float* __restrict__ xhat, float* __restrict__ partial) {
  __shared__ int   sIdx[TOPK];
  __shared__ float sVal[TOPK];
  __shared__ float sred[256];
  const int row = blockIdx.x;
  const int tid = threadIdx.x;
  const int n = cnt[row];
  if (tid < TOPK) {
    sIdx[tid] = idxl[(size_t)row * TOPK + tid];
    sVal[tid] = vall[(size_t)row * TOPK + tid];
  }
  __syncthreads();

  float acc[18];                         // 4608 = 18 * 256
  #pragma unroll
  for (int k = 0; k < 18; k++) acc[k] = 0.0f;

  for (int j = 0; j < n; j++) {
    const float* wrow = W_dec + (size_t)sIdx[j] * ROW_DEC;
    const float v = sVal[j];
    #pragma unroll
    for (int k = 0; k < 18; k++)
      acc[k] = fmaf(v, wrow[tid + k * 256], acc[k]);
  }

  float sse = 0.0f;
  #pragma unroll
  for (int k = 0; k < 18; k++) {
    const int pos = tid + k * 256;
    const float xh = acc[k] + b_dec[pos];
    xhat[(size_t)row * ROW_DEC + pos] = xh;
    const float d = xh - x[(size_t)row * ROW_DEC + pos];
    sse = fmaf(d, d, sse);
  }
  sred[tid] = sse;
  __syncthreads();
  for (int s = 128; s > 0; s >>= 1) {
    if (tid < s) sred[tid] += sred[tid + s];
    __syncthreads();
  }
  if (tid == 0) partial[row] = sred[0];
}

// ---- K4: deterministic final reduction -------------------------------------
__global__ __launch_bounds__(256) void sae_reduce_kernel(
    const float* __restrict__ partial, float* __restrict__ out0) {
  __shared__ float sred[256];
  const int tid = threadIdx.x;
  float s = 0.0f;
  for (int i = tid; i < B_ROWS; i += 256) s += partial[i];
  sred[tid] = s;
  __syncthreads();
  for (int st = 128; st > 0; st >>= 1) {
    if (tid < st) sred[tid] += sred[tid + st];
    __syncthreads();
  }
  if (tid == 0) out0[0] = sred[0] / (float)(B_ROWS * 6);   // mean over B*T*L
}

// ---- host-side launcher -----------------------------------------------------
extern "C" void kernel_launch(void* const* d_in, const int* in_sizes, int n_in,
                              void* d_out, int out_size, void* d_ws, size_t ws_size,
                              hipStream_t stream) {
  const float* x     = (const float*)d_in[0];
  const float* W_enc = (const float*)d_in[1];
  const float* b_enc = (const float*)d_in[2];
  const float* W_dec = (const float*)d_in[3];
  const float* b_dec = (const float*)d_in[4];

  float* out  = (float*)d_out;
  float* xhat = out + 1;
  float* zbuf = out + 1 + (size_t)B_ROWS * ROW_DEC;   // z region; used as pre-scratch

  char* ws = (char*)d_ws;
  size_t off = 0;
  float* xsum    = (float*)(ws + off); off += (size_t)B_ROWS * KDIM * sizeof(float);
  int*   cnt     = (int*)  (ws + off); off += (size_t)B_ROWS * sizeof(int);
  int*   idxl    = (int*)  (ws + off); off += (size_t)B_ROWS * TOPK * sizeof(int);
  float* vall    = (float*)(ws + off); off += (size_t)B_ROWS * TOPK * sizeof(float);
  float* partial = (float*)(ws + off); off += (size_t)B_ROWS * sizeof(float);

  // K0: xsum
  sae_xsum_kernel<<<(B_ROWS * KDIM) / (256 * 4), 256, 0, stream>>>(x, xsum);

  // K1: encode GEMM (WMMA f32 + TDM double buffering), pre -> z region
  dim3 ggrid(D_SAE / BN, B_ROWS / BM);                 // (64, 128)
  sae_enc_gemm_kernel<<<ggrid, 256, GEMM_LDS, stream>>>(xsum, W_enc, b_enc, zbuf);

  // K2: top-k + relu + compaction
  const size_t dynLds = D_SAE * sizeof(float) + 512 * sizeof(int);
  sae_topk_kernel<<<B_ROWS, 256, dynLds, stream>>>(zbuf, cnt, idxl, vall);

  // K3: sparse decode + SSE partials
  sae_decode_kernel<<<B_ROWS, 256, 0, stream>>>(x, W_dec, b_dec, cnt, idxl, vall,
                                                xhat, partial);

  // K4: final loss
  sae_reduce_kernel<<<1, 256, 0, stream>>>(partial, out);
}